// GLBLPathwayMLP_SingleLayer_33981781246565
// MI455X (gfx1250) — compile-verified
//
#include <hip/hip_runtime.h>
#include <math.h>

#define B_  8192
#define D_  784
#define H_  2048
#define O_  1024
#define RH_ 128
#define P_  64
#define BT  16      // batch rows per block
#define WHS 516     // padded row stride of wh tile (bank-conflict-free)

typedef float v2f __attribute__((ext_vector_type(2)));
typedef float v8f __attribute__((ext_vector_type(8)));
typedef int   v4i32 __attribute__((vector_size(16)));

#if defined(__has_builtin)
#  if __has_builtin(__builtin_amdgcn_global_load_async_to_lds_b128)
#    define USE_ASYNC_LDS 1
#  endif
#endif
#ifndef USE_ASYNC_LDS
#  define USE_ASYNC_LDS 0
#endif

// Copy 16 bytes global -> LDS (async CDNA5 path when available)
__device__ __forceinline__ void cp16_to_lds(float* ldst, const float* gsrc) {
#if USE_ASYNC_LDS
  typedef __attribute__((address_space(1))) v4i32 gv4_t;
  typedef __attribute__((address_space(3))) v4i32 lv4_t;
  __builtin_amdgcn_global_load_async_to_lds_b128((gv4_t*)gsrc, (lv4_t*)ldst, 0, 0);
#else
  *(float4*)ldst = *(const float4*)gsrc;
#endif
}

__device__ __forceinline__ void lds_copy_fence() {
  asm volatile("s_wait_asynccnt 0" ::: "memory");
}

// Branch-free gelu(x) = 0.5 x (1 + erf(x/sqrt(2))), A&S 7.1.26 erf (|abs err| <= 1.5e-7)
__device__ __forceinline__ float gelu_f(float x) {
  float z  = x * 0.70710678118654752f;
  float az = fabsf(z);
  float t  = __builtin_amdgcn_rcpf(fmaf(0.3275911f, az, 1.0f));
  float p  = fmaf(1.061405429f, t, -1.453152027f);
  p = fmaf(p, t, 1.421413741f);
  p = fmaf(p, t, -0.284496736f);
  p = fmaf(p, t, 0.254829592f);
  p = p * t;
  float e  = __expf(-az * az);
  float er = copysignf(fmaf(-p, e, 1.0f), z);   // erf(z)
  return 0.5f * x * (1.0f + er);
}

// W1P[(i*196+n)][h] = W1[h][ig[i*196+n]]  (one-time gather+transpose; coalesced writes)
__global__ __launch_bounds__(256) void permuteW1_k(const float* __restrict__ W1,
                                                   const int* __restrict__ ig,
                                                   float* __restrict__ W1P) {
  int idx = blockIdx.x * blockDim.x + threadIdx.x;
  if (idx < D_ * H_) {
    int row = idx >> 11, h = idx & (H_ - 1);
    W1P[idx] = W1[(size_t)h * D_ + ig[row]];
  }
}

// W2T[h][o] = W2[o][h]  (one-time transpose; coalesced writes)
__global__ __launch_bounds__(256) void transposeW2_k(const float* __restrict__ W2,
                                                     float* __restrict__ W2T) {
  int idx = blockIdx.x * blockDim.x + threadIdx.x;
  if (idx < H_ * O_) {
    int h = idx >> 10, o = idx & (O_ - 1);
    W2T[idx] = W2[(size_t)o * H_ + h];
  }
}

// xp[b][c] = x[b][ig[c]]  (one-time column permutation of x into group order)
__global__ __launch_bounds__(256) void permuteX_k(const float* __restrict__ x,
                                                  const int* __restrict__ ig,
                                                  float* __restrict__ xp) {
  int idx = blockIdx.x * blockDim.x + threadIdx.x;
  if (idx < B_ * D_) {
    int b = idx / D_, c = idx - b * D_;
    xp[idx] = x[(size_t)b * D_ + ig[c]];
  }
}

// Router: h_r = relu(x Wr1^T + br1); scores = h_r Wr2^T + br2; softmax; top-8; renorm.
__global__ __launch_bounds__(128) void router_k(const float* __restrict__ x,
                                                const float* __restrict__ Wr1,
                                                const float* __restrict__ br1,
                                                const float* __restrict__ Wr2,
                                                const float* __restrict__ br2,
                                                float* __restrict__ wts) {
  __shared__ float xrow[D_];
  __shared__ float hr[RH_];
  __shared__ float pr[P_];
  __shared__ float wout[P_];
  __shared__ float invs;
  int row = blockIdx.x, t = threadIdx.x;
  const float* xr = x + (size_t)row * D_;
  for (int d = t; d < D_; d += 128) xrow[d] = xr[d];
  __syncthreads();
  {
    const float* w = Wr1 + (size_t)t * D_;
    float acc = br1[t];
    for (int d = 0; d < D_; ++d) acc = fmaf(xrow[d], w[d], acc);
    hr[t] = fmaxf(acc, 0.0f);
  }
  __syncthreads();
  if (t < P_) {
    const float* w = Wr2 + t * RH_;
    float acc = br2[t];
    for (int d = 0; d < RH_; ++d) acc = fmaf(hr[d], w[d], acc);
    pr[t] = acc;
    wout[t] = 0.0f;
  }
  __syncthreads();
  if (t == 0) {
    float mx = pr[0];
    for (int p = 1; p < P_; ++p) mx = fmaxf(mx, pr[p]);
    float s = 0.0f;
    for (int p = 0; p < P_; ++p) { float e = __expf(pr[p] - mx); pr[p] = e; s += e; }
    float inv = 1.0f / s;
    for (int p = 0; p < P_; ++p) pr[p] *= inv;
    float ssum = 0.0f;
    for (int it = 0; it < 8; ++it) {      // top-8 (first occurrence on ties, like lax.top_k)
      int bi = 0; float bv = -1.0f;
      for (int p = 0; p < P_; ++p)
        if (wout[p] == 0.0f && pr[p] > bv) { bv = pr[p]; bi = p; }
      wout[bi] = bv; ssum += bv;
    }
    invs = 1.0f / (ssum + 1e-8f);
  }
  __syncthreads();
  if (t < P_) wts[(size_t)row * P_ + t] = wout[t] * invs;
}

// Fused main path. Block = 256 threads (8 waves), BT=16 batch rows.
// Wave w owns hidden cols [w*64, w*64+64) for GEMM1 and output cols [w*128, w*128+128) (k-block w/2).
__global__ __launch_bounds__(256) void pathway_k(
    const float* __restrict__ xp, const float* __restrict__ W1P,
    const float* __restrict__ b1, const float* __restrict__ W2T,
    const float* __restrict__ b2, const float* __restrict__ wts,
    float* __restrict__ out) {
  extern __shared__ float lds[];
  float* xg  = lds;                       // [BT][196]  staged x_reg group tile
  float* wh  = lds + BT * 196;            // [4][BT][WHS] k-weighted hidden (per j)
  float* wtl = wh + 4 * BT * WHS;         // [BT][64]   router weights tile

  const int tid   = threadIdx.x;
  const int wave  = tid >> 5;
  const int lane  = tid & 31;
  const int laneN = lane & 15;
  const int khalf = lane >> 4;
  const int rowbase = blockIdx.x * BT;
  const int kb    = wave >> 1;            // output k-block this wave serves
  const int obase = wave * 128;
  const int hwb   = wave * 64;            // hidden-col base owned by this wave

  for (int idx = tid; idx < BT * 64; idx += 256) {
    int m = idx >> 6, p = idx & 63;
    wtl[idx] = wts[(size_t)(rowbase + m) * 64 + p];
  }

  v8f outacc[8];
  #pragma unroll
  for (int t = 0; t < 8; ++t)
    #pragma unroll
    for (int q = 0; q < 8; ++q) outacc[t][q] = 0.0f;

  for (int j = 0; j < 4; ++j) {
    // bias for this wave's hidden cols (per j): 4 values per lane
    float bias_t[4];
    #pragma unroll
    for (int t = 0; t < 4; ++t) bias_t[t] = b1[j * 512 + hwb + t * 16 + laneN];

    // k-weighted hidden accumulators, held in registers across the i loop
    v8f wacc[4][4];                       // [k][tile]
    #pragma unroll
    for (int k = 0; k < 4; ++k)
      #pragma unroll
      for (int t = 0; t < 4; ++t)
        #pragma unroll
        for (int q = 0; q < 8; ++q) wacc[k][t][q] = 0.0f;

    for (int i = 0; i < 4; ++i) {
      __syncthreads();                    // xg reuse + (i==0) prev-j GEMM2 readers done
      // Stage x_reg tile: contiguous 16B chunks from pre-permuted xp (async->LDS path)
      for (int idx = tid; idx < BT * 49; idx += 256) {
        int m = idx / 49, q = idx - m * 49;
        cp16_to_lds(xg + m * 196 + q * 4,
                    xp + (size_t)(rowbase + m) * D_ + i * 196 + q * 4);
      }
      lds_copy_fence();
      __syncthreads();

      // GEMM1: hidden(16x512) tiles for this wave's 64 cols, fp32 WMMA, K-steps of 4
      v8f hacc[4];
      #pragma unroll
      for (int t = 0; t < 4; ++t)
        #pragma unroll
        for (int q = 0; q < 8; ++q) hacc[t][q] = 0.0f;

      const float* ap = xg + laneN * 196 + 2 * khalf;
      const float* bp = W1P + (size_t)(i * 196 + 2 * khalf) * H_ + j * 512 + hwb + laneN;
      #pragma unroll 1   // single body: keeps accumulators pinned, no PHI-copy ping-pong
      for (int kk = 0; kk < 196; kk += 4) {
        v2f a;                            // A[m][K], m=laneN, K=2*khalf+v
        a[0] = ap[kk]; a[1] = ap[kk + 1];
        #pragma unroll
        for (int t = 0; t < 4; ++t) {
          v2f bf;                         // B[K][n]; linear addresses, no gathers
          bf[0] = bp[(size_t)kk * H_ + t * 16];
          bf[1] = bp[(size_t)(kk + 1) * H_ + t * 16];
          hacc[t] = __builtin_amdgcn_wmma_f32_16x16x4_f32(
              false, a, false, bf, (short)0, hacc[t], false, false);
        }
      }

      // bias + gelu + weighted accumulate into registers (no LDS RMW)
      #pragma unroll
      for (int v = 0; v < 8; ++v) {
        int m = v + 8 * khalf;            // C-layout row
        float4 wv4 = *(const float4*)&wtl[m * 64 + i * 16 + j * 4];
        #pragma unroll
        for (int t = 0; t < 4; ++t) {
          float g = gelu_f(hacc[t][v] + bias_t[t]);
          wacc[0][t][v] = fmaf(g, wv4.x, wacc[0][t][v]);
          wacc[1][t][v] = fmaf(g, wv4.y, wacc[1][t][v]);
          wacc[2][t][v] = fmaf(g, wv4.z, wacc[2][t][v]);
          wacc[3][t][v] = fmaf(g, wv4.w, wacc[3][t][v]);
        }
      }
    } // i

    // single store pass: wacc -> wh (pure ds_store, exclusive column ownership)
    #pragma unroll
    for (int k = 0; k < 4; ++k)
      #pragma unroll
      for (int t = 0; t < 4; ++t) {
        int hcl = hwb + t * 16 + laneN;
        #pragma unroll
        for (int v = 0; v < 8; ++v) {
          int m = v + 8 * khalf;
          wh[(k * BT + m) * WHS + hcl] = wacc[k][t][v];
        }
      }

    __syncthreads();                      // wh complete for this j
    // GEMM2 partial: out[16 x 128] += wh[kb] (16x512) * W2T_j (512x128 slice)
    const float* ap2 = wh + (kb * BT + laneN) * WHS + 2 * khalf;
    const float* bp2 = W2T + (size_t)(j * 512 + 2 * khalf) * O_ + obase + laneN;
    #pragma unroll 1
    for (int hb = 0; hb < 512; hb += 4) {
      v2f a;
      a[0] = ap2[hb]; a[1] = ap2[hb + 1];
      #pragma unroll
      for (int t = 0; t < 8; ++t) {
        v2f bf;
        bf[0] = bp2[(size_t)hb * O_ + t * 16];
        bf[1] = bp2[(size_t)(hb + 1) * O_ + t * 16];
        outacc[t] = __builtin_amdgcn_wmma_f32_16x16x4_f32(
            false, a, false, bf, (short)0, outacc[t], false, false);
      }
    }
  } // j

  // finalize: out += wsum[b, kb] * b2
  float wsum_v[8];
  #pragma unroll
  for (int v = 0; v < 8; ++v) {
    int m = v + 8 * khalf;
    float s = 0.0f;
    #pragma unroll
    for (int p = 0; p < 16; ++p) s += wtl[m * 64 + p * 4 + kb];
    wsum_v[v] = s;
  }
  #pragma unroll
  for (int t = 0; t < 8; ++t) {
    int oc = obase + t * 16 + laneN;
    float b2v = b2[oc];
    #pragma unroll
    for (int v = 0; v < 8; ++v) {
      int m = v + 8 * khalf;
      out[(size_t)(rowbase + m) * O_ + oc] = outacc[t][v] + wsum_v[v] * b2v;
    }
  }
}

extern "C" void kernel_launch(void* const* d_in, const int* in_sizes, int n_in,
                              void* d_out, int out_size, void* d_ws, size_t ws_size,
                              hipStream_t stream) {
  (void)in_sizes; (void)n_in; (void)out_size; (void)ws_size;
  const float* x   = (const float*)d_in[0];
  const float* W1  = (const float*)d_in[1];
  const float* b1  = (const float*)d_in[2];
  const float* W2  = (const float*)d_in[3];
  const float* b2  = (const float*)d_in[4];
  const float* Wr1 = (const float*)d_in[5];
  const float* br1 = (const float*)d_in[6];
  const float* Wr2 = (const float*)d_in[7];
  const float* br2 = (const float*)d_in[8];
  const int*   ig  = (const int*)d_in[9];
  float* out = (float*)d_out;

  float* ws  = (float*)d_ws;
  float* wts = ws;                                  // B * 64
  float* W1P = wts + (size_t)B_ * P_;               // 784 * 2048 (gathered+transposed W1)
  float* W2T = W1P + (size_t)D_ * H_;               // 2048 * 1024
  float* xp  = W2T + (size_t)H_ * O_;               // B * 784   (group-permuted x)

  permuteW1_k<<<(D_ * H_ + 255) / 256, 256, 0, stream>>>(W1, ig, W1P);
  transposeW2_k<<<(H_ * O_ + 255) / 256, 256, 0, stream>>>(W2, W2T);
  permuteX_k<<<(B_ * D_ + 255) / 256, 256, 0, stream>>>(x, ig, xp);
  router_k<<<B_, 128, 0, stream>>>(x, Wr1, br1, Wr2, br2, wts);

  size_t shmem = (size_t)(BT * 196 + 4 * BT * WHS + BT * 64) * sizeof(float);
  pathway_k<<<B_ / BT, 256, shmem, stream>>>(xp, W1P, b1, W2T, b2, wts, out);
}